// _HashGridMLP_41180146434627
// MI455X (gfx1250) — compile-verified
//
#include <hip/hip_runtime.h>
#include <hip/hip_bf16.h>
#include <stdint.h>

// ext_vector float4: real clang vector type -> elementwise ops + works with
// __builtin_nontemporal_store, lowers to b128 loads/stores.
typedef float v4f __attribute__((ext_vector_type(4)));

#define TPB 256
#define HASH_MASK 0x7FFFFu          // HASHMAP_SIZE = 2^19
#define PRIME1 2654435761u
#define PRIME2 805459861u
#define RES 128.0f

__global__ __launch_bounds__(TPB)
void _HashGrid_gather_kernel(const float* __restrict__ x,
                             const float* __restrict__ emb,
                             float* __restrict__ out,
                             int npts)
{
    __shared__ float sx[TPB * 3];

    const int tid = threadIdx.x;
    const long long base  = (long long)blockIdx.x * TPB;   // first point of tile
    const long long base3 = base * 3;                      // first float of tile
    const long long lastElem = (long long)npts * 3 - 1;

    // ---- CDNA5 async copy: stage x tile (TPB*3 floats) into LDS ----------
    // Flat LDS pointers have the LDS byte offset in their low 32 bits
    // (aperture addr[63:32] is the SHARED_BASE tag), so truncation gives the
    // per-lane VDST LDS address for global_load_async_to_lds_b32.
    unsigned lds_base = (unsigned)(uintptr_t)(&sx[0]);
#pragma unroll
    for (int k = 0; k < 3; ++k) {
        long long ge = base3 + tid + k * TPB;
        if (ge > lastElem) ge = lastElem;      // clamp tail: no divergence, in-bounds
        unsigned lds_addr = lds_base + (unsigned)(tid + k * TPB) * 4u;
        const float* gp = x + ge;
        asm volatile("global_load_async_to_lds_b32 %0, %1, off"
                     :: "v"(lds_addr), "v"(gp)
                     : "memory");
    }
    asm volatile("s_wait_asynccnt 0" ::: "memory");  // our wave's copies landed
    __syncthreads();                                 // all waves' copies visible

    // ---- per-point setup --------------------------------------------------
    float xr0 = sx[3 * tid + 0] * RES;
    float xr1 = sx[3 * tid + 1] * RES;
    float xr2 = sx[3 * tid + 2] * RES;
    float fl0 = floorf(xr0), fl1 = floorf(xr1), fl2 = floorf(xr2);
    float xf0 = xr0 - fl0,   xf1 = xr1 - fl1,   xf2 = xr2 - fl2;
    unsigned xi0 = (unsigned)(int)fl0;
    unsigned xi1 = (unsigned)(int)fl1;
    unsigned xi2 = (unsigned)(int)fl2;

    // hash partials per dim (u32 wraparound multiply), corner bit d==0 -> xi,
    // bit d==1 -> xi+1 (matches BIN_MASK: mask true when bit clear)
    unsigned ha0 = xi0;                  unsigned hb0 = xi0 + 1u;
    unsigned ha1 = xi1 * PRIME1;         unsigned hb1 = ha1 + PRIME1;
    unsigned ha2 = xi2 * PRIME2;         unsigned hb2 = ha2 + PRIME2;
    // weight partials: bit clear -> (1-xf), bit set -> xf
    float wa0 = 1.0f - xf0, wb0 = xf0;
    float wa1 = 1.0f - xf1, wb1 = xf1;
    float wa2 = 1.0f - xf2, wb2 = xf2;

    // ---- issue all 16 b128 gathers first (max loads in flight) -----------
    const v4f* embv = (const v4f*)emb;   // 2 x v4f per 8-float row, 32B aligned
    v4f f[16];
    float w[8];
#pragma unroll
    for (int c = 0; c < 8; ++c) {
        unsigned h = ((c & 1) ? hb0 : ha0)
                   ^ ((c & 2) ? hb1 : ha1)
                   ^ ((c & 4) ? hb2 : ha2);
        h &= HASH_MASK;
        w[c] = ((c & 1) ? wb0 : wa0)
             * ((c & 2) ? wb1 : wa1)
             * ((c & 4) ? wb2 : wa2);
        const v4f* row = embv + (size_t)h * 2u;
        f[2 * c + 0] = row[0];
        f[2 * c + 1] = row[1];
    }

    // ---- trilinear weighted accumulation ----------------------------------
    v4f acc0 = {0.f, 0.f, 0.f, 0.f};
    v4f acc1 = {0.f, 0.f, 0.f, 0.f};
#pragma unroll
    for (int c = 0; c < 8; ++c) {
        acc0 += w[c] * f[2 * c + 0];
        acc1 += w[c] * f[2 * c + 1];
    }

    // ---- non-temporal streaming store: keep L2 for the embedding table ----
    long long i = base + tid;
    if (i < npts) {
        v4f* o = (v4f*)out + (size_t)i * 2u;
        __builtin_nontemporal_store(acc0, o + 0);
        __builtin_nontemporal_store(acc1, o + 1);
    }
}

extern "C" void kernel_launch(void* const* d_in, const int* in_sizes, int n_in,
                              void* d_out, int out_size, void* d_ws, size_t ws_size,
                              hipStream_t stream) {
    const float* x   = (const float*)d_in[0];   // [N, 3] f32
    const float* emb = (const float*)d_in[1];   // [524288, 8] f32
    float* out = (float*)d_out;                 // [N, 8] f32
    int npts = in_sizes[0] / 3;
    int grid = (npts + TPB - 1) / TPB;
    hipLaunchKernelGGL(_HashGrid_gather_kernel, dim3(grid), dim3(TPB), 0, stream,
                       x, emb, out, npts);
}